// GraphAttentionLayer_44315472560709
// MI455X (gfx1250) — compile-verified
//
#include <hip/hip_runtime.h>
#include <math.h>

// Problem constants from the reference
#define Bb 2
#define Tt 16
#define Nn 256
#define Dd 64            // D_IN == D_OUT == 64
#define ROWS (Bb*Tt*Nn)  // 8192
#define ALPHA 0.2f

typedef __attribute__((ext_vector_type(2))) float v2f;
typedef __attribute__((ext_vector_type(8))) float v8f;

// ---------------------------------------------------------------------------
// Kernel A: h = x @ W1 + b1  (f32 WMMA 16x16x4), plus s_src = h.a_src,
// s_dst = h.a_dst.  4 waves per block; each wave owns a 16x64 tile of h.
// ---------------------------------------------------------------------------
__global__ __launch_bounds__(128)
void gat_h_wmma_kernel(const float* __restrict__ x,      // [ROWS,64]
                       const float* __restrict__ W1,     // [64,64]
                       const float* __restrict__ b1,     // [64]
                       const float* __restrict__ a_src,  // [64]
                       const float* __restrict__ a_dst,  // [64]
                       float* __restrict__ h,            // [ROWS,64]
                       float* __restrict__ s_src,        // [ROWS]
                       float* __restrict__ s_dst)        // [ROWS]
{
    __shared__ float lh[64][65];   // block's 64 rows of h, padded pitch

    const int wave = threadIdx.x >> 5;
    const int lane = threadIdx.x & 31;
    const int row0 = blockIdx.x * 64 + wave * 16;   // wave's first row

    // WMMA f32 16x16x4 lane mapping:
    //   A (16x4): lane = m + 16*(k>=2); VGPR0 -> k = kb, VGPR1 -> k = kb+1
    //   B (4x16): lane = n + 16*(k>=2); same k split
    const int mn   = lane & 15;          // m for A, n for B
    const int kb   = (lane >> 4) * 2;    // 0 or 2
    const int half = lane >> 4;

    v8f acc[4];
#pragma unroll
    for (int nt = 0; nt < 4; ++nt) acc[nt] = (v8f){0,0,0,0,0,0,0,0};

#pragma unroll
    for (int ks = 0; ks < 16; ++ks) {
        const int k = ks * 4 + kb;
        v2f a;
        const float* xr = x + (size_t)(row0 + mn) * Dd + k;
        a.x = xr[0];
        a.y = xr[1];
#pragma unroll
        for (int nt = 0; nt < 4; ++nt) {
            v2f bf;
            bf.x = W1[(size_t)k       * Dd + nt * 16 + mn];
            bf.y = W1[(size_t)(k + 1) * Dd + nt * 16 + mn];
            acc[nt] = __builtin_amdgcn_wmma_f32_16x16x4_f32(
                false, a, false, bf, (short)0, acc[nt], false, false);
        }
    }

    // C/D layout: lane n = lane&15 ; VGPR j holds rows m = j + 8*(lane>>4)
#pragma unroll
    for (int nt = 0; nt < 4; ++nt) {
        const float bias = b1[nt * 16 + mn];
#pragma unroll
        for (int j = 0; j < 8; ++j) {
            const int rm = j + 8 * half;                 // row within tile
            const float v = acc[nt][j] + bias;
            lh[wave * 16 + rm][nt * 16 + mn] = v;
            h[(size_t)(row0 + rm) * Dd + nt * 16 + mn] = v;
        }
    }
    __syncthreads();

    // s_src / s_dst: one thread per block-row
    if (threadIdx.x < 64) {
        const int r = threadIdx.x;
        float ss = 0.f, sd = 0.f;
#pragma unroll
        for (int d = 0; d < Dd; ++d) {
            const float v = lh[r][d];
            ss += v * a_src[d];
            sd += v * a_dst[d];
        }
        const int grow = blockIdx.x * 64 + r;
        s_src[grow] = ss;
        s_dst[grow] = sd;
    }
}

// ---------------------------------------------------------------------------
// Kernel B: attention diagonal + output.  One block per (b,n); 256 threads,
// thread tid owns column m = tid.
// ---------------------------------------------------------------------------
__global__ __launch_bounds__(256)
void gat_att_kernel(const float* __restrict__ h,      // [ROWS,64]
                    const float* __restrict__ s_src,  // [ROWS]
                    const float* __restrict__ s_dst,  // [ROWS]
                    const int*   __restrict__ adj,    // [B,T,N,N]
                    const float* __restrict__ b2p,    // [1]
                    float* __restrict__ out)          // [B,T,N,64]
{
    const int bn = blockIdx.x;        // 0..511
    const int b  = bn >> 8;
    const int n  = bn & 255;
    const int m  = threadIdx.x;       // 0..255
    const float b2 = b2p[0];

    __shared__ float ssrc[Tt];
    __shared__ float wmax[8];
    __shared__ float ediag[Tt];
    __shared__ float wpart[8][Tt];
    __shared__ float diag_s[Tt];

    if (threadIdx.x < Tt)
        ssrc[threadIdx.x] = s_src[(size_t)(b * Tt + threadIdx.x) * Nn + n];
    __syncthreads();

    // e[t,m] for this column, local max over t
    float vals[Tt];
    float lmax = -INFINITY;
#pragma unroll
    for (int t = 0; t < Tt; ++t) {
        float v = ssrc[t] + s_dst[(size_t)(b * Tt + t) * Nn + m] + b2;
        v = v > 0.f ? v : ALPHA * v;          // LeakyReLU
        vals[t] = v;
        lmax = fmaxf(lmax, v);
    }
    // block-wide max (wave32 shuffle reduce, then LDS across 8 waves)
#pragma unroll
    for (int off = 16; off >= 1; off >>= 1)
        lmax = fmaxf(lmax, __shfl_xor(lmax, off, 32));
    if ((threadIdx.x & 31) == 0) wmax[threadIdx.x >> 5] = lmax;
    __syncthreads();
    float M = wmax[0];
#pragma unroll
    for (int w = 1; w < 8; ++w) M = fmaxf(M, wmax[w]);

    // column sums and diagonal numerators
    float csum = 0.f;
#pragma unroll
    for (int t = 0; t < Tt; ++t) {
        vals[t] = expf(vals[t] - M);
        csum += vals[t];
    }
    if (m == n) {
#pragma unroll
        for (int r = 0; r < Tt; ++r) ediag[r] = vals[r];
    }

    // denominators: denom[r] = sum_m [adj[b,r,n,m]!=0] * csum[m]
    const size_t adjbase = (((size_t)b * Tt) * Nn + n) * Nn + m;  // +r*N*N
#pragma unroll
    for (int r = 0; r < Tt; ++r) {
        float c = (adj[adjbase + (size_t)r * Nn * Nn] != 0) ? csum : 0.f;
#pragma unroll
        for (int off = 16; off >= 1; off >>= 1)
            c += __shfl_xor(c, off, 32);
        if ((threadIdx.x & 31) == 0) wpart[threadIdx.x >> 5][r] = c;
    }
    __syncthreads();

    if (threadIdx.x < Tt) {
        const int r = threadIdx.x;
        float dn = 0.f;
#pragma unroll
        for (int w = 0; w < 8; ++w) dn += wpart[w][r];
        const int adjdiag = adj[(((size_t)(b * Tt + r)) * Nn + n) * Nn + n];
        float dg;
        if (dn == 0.f)       dg = 1.0f / (float)(Tt * Nn);  // all-masked row
        else if (adjdiag)    dg = ediag[r] / dn;
        else                 dg = 0.f;                      // masked diagonal
        diag_s[r] = dg;
    }
    __syncthreads();

    // out[b,r,n,d] = elu(diag[r] * h[b,r,n,d])
    for (int i = threadIdx.x; i < Tt * Dd; i += 256) {
        const int r = i >> 6;
        const int d = i & 63;
        const size_t idx = (((size_t)(b * Tt + r)) * Nn + n) * Dd + d;
        const float v = diag_s[r] * h[idx];
        out[idx] = v > 0.f ? v : (expf(v) - 1.f);
    }
}

// ---------------------------------------------------------------------------
extern "C" void kernel_launch(void* const* d_in, const int* in_sizes, int n_in,
                              void* d_out, int out_size, void* d_ws, size_t ws_size,
                              hipStream_t stream) {
    const float* x     = (const float*)d_in[0];
    const int*   adj   = (const int*)  d_in[1];
    const float* W1    = (const float*)d_in[2];
    const float* b1    = (const float*)d_in[3];
    const float* a_src = (const float*)d_in[4];
    const float* a_dst = (const float*)d_in[5];
    const float* b2    = (const float*)d_in[6];
    float* out = (float*)d_out;

    float* h     = (float*)d_ws;             // ROWS*64 floats
    float* s_src = h + (size_t)ROWS * Dd;    // ROWS floats
    float* s_dst = s_src + ROWS;             // ROWS floats

    gat_h_wmma_kernel<<<ROWS / 64, 128, 0, stream>>>(x, W1, b1, a_src, a_dst,
                                                     h, s_src, s_dst);
    gat_att_kernel<<<Bb * Nn, 256, 0, stream>>>(h, s_src, s_dst, adj, b2, out);
}